// BranchBlock_59416577573014
// MI455X (gfx1250) — compile-verified
//
#include <hip/hip_runtime.h>
#include <hip/hip_bf16.h>

typedef __attribute__((ext_vector_type(2))) float v2f;
typedef __attribute__((ext_vector_type(8))) float v8f;

// ---------------------------------------------------------------------------
// wave-local LDS ordering: CDNA5 split dep counters (s_wait_dscnt) + compiler
// scheduling fence. LDS ops from one wave are processed in order at the LDS,
// so this is sufficient for intra-wave producer/consumer through LDS.
// ---------------------------------------------------------------------------
__device__ __forceinline__ void wave_lds_sync() {
  __builtin_amdgcn_wave_barrier();
  asm volatile("s_wait_dscnt 0" ::: "memory");
  __builtin_amdgcn_wave_barrier();
}

// CDNA5 async copy global->LDS (ASYNCcnt-tracked), 16B per lane.
// LDS destination offset = low 32 bits of the generic pointer (LDS aperture
// maps flat addr[31:0] to the LDS offset).
__device__ __forceinline__ void async_load_b128(uint32_t lds_off, const void* gptr) {
  asm volatile("global_load_async_to_lds_b128 %0, %1, off"
               :: "v"(lds_off), "v"(gptr) : "memory");
}
__device__ __forceinline__ void wait_asynccnt0() {
  asm volatile("s_wait_asynccnt 0x0" ::: "memory");
}

__device__ __forceinline__ int lower_bound_i32(const int* __restrict__ a, int n, int v) {
  int lo = 0, hi = n;
  while (lo < hi) { int mid = (lo + hi) >> 1; if (a[mid] < v) lo = mid + 1; else hi = mid; }
  return lo;
}

// ---------------------------------------------------------------------------
// BatchNorm batch statistics: one block per channel, biased variance, eps=1e-5
// ---------------------------------------------------------------------------
__global__ __launch_bounds__(256) void bn_stats_kernel(
    const float* __restrict__ X, int rows, int C,
    float* __restrict__ mean, float* __restrict__ rstd) {
  __shared__ float sb[256];
  __shared__ float sb2[256];
  int c = blockIdx.x;
  float s = 0.f, s2 = 0.f;
  for (int r = threadIdx.x; r < rows; r += blockDim.x) {
    float v = X[(size_t)r * C + c];
    s += v; s2 += v * v;
  }
  sb[threadIdx.x] = s; sb2[threadIdx.x] = s2;
  __syncthreads();
  for (int off = 128; off > 0; off >>= 1) {
    if ((int)threadIdx.x < off) {
      sb[threadIdx.x]  += sb[threadIdx.x + off];
      sb2[threadIdx.x] += sb2[threadIdx.x + off];
    }
    __syncthreads();
  }
  if (threadIdx.x == 0) {
    float m   = sb[0] / (float)rows;
    float var = sb2[0] / (float)rows - m * m;
    mean[c] = m;
    rstd[c] = rsqrtf(var + 1e-5f);
  }
}

// ---------------------------------------------------------------------------
// Y = leaky_relu((X - mean) * rstd * g + b, slope)
// ---------------------------------------------------------------------------
__global__ __launch_bounds__(256) void bn_act_kernel(
    const float* __restrict__ X, const float* __restrict__ mean,
    const float* __restrict__ rstd, const float* __restrict__ g,
    const float* __restrict__ b, float slope,
    float* __restrict__ Y, int total, int C) {
  int i = blockIdx.x * blockDim.x + threadIdx.x;
  if (i >= total) return;
  int c = i % C;
  float v = (X[i] - mean[c]) * rstd[c] * g[c] + b[c];
  Y[i] = v > 0.f ? v : slope * v;
}

// ---------------------------------------------------------------------------
// C[M,N] = A[M,K] @ B[K,N] (+ Cinit), f32 WMMA 16x16x4.
// Block = 4 waves covering a 64x16 output stripe sharing one B panel:
// the K x 16 panel is staged into LDS with global_load_async_to_lds_b128
// (ASYNCcnt) and reused by all 4 waves (4x reduction of B traffic).
// A layout (ISA 7.12.2, 32-bit A 16x4): lane&15 = M, lane>>4 selects K pair.
// C/D layout: VGPR v -> row (lane>>4)*8 + v, col = lane&15.
// ---------------------------------------------------------------------------
__global__ __launch_bounds__(128) void gemm_wmma_f32(
    const float* __restrict__ A, const float* __restrict__ B,
    const float* __restrict__ Cinit, float* __restrict__ C,
    int M, int N, int K) {
  __shared__ float panelS[128 * 16];      // up to K=128 rows of 16 floats (8KB)

  int lane   = threadIdx.x & 31;
  int wv     = threadIdx.x >> 5;
  int tilesN = N >> 4;
  int bRow   = blockIdx.x / tilesN;
  int bCol   = blockIdx.x % tilesN;
  int row0   = bRow * 64 + wv * 16;
  int col0   = bCol * 16;
  if (row0 >= M) return;                  // block-uniform with exact tiling

  // cooperative async stage of the B[K x 16] panel: 16B per lane per round
  for (int r0 = 0; r0 < K; r0 += 32) {
    int r = r0 + ((int)threadIdx.x >> 2);
    int q = ((int)threadIdx.x & 3) << 2;  // float quad within row
    async_load_b128((uint32_t)(uintptr_t)&panelS[r * 16 + q],
                    B + (size_t)r * N + col0 + q);
  }
  wait_asynccnt0();
  __syncthreads();

  int m    = lane & 15;
  int half = lane >> 4;
  int kk   = half << 1;

  v8f acc = {0.f, 0.f, 0.f, 0.f, 0.f, 0.f, 0.f, 0.f};
  if (Cinit) {
#pragma unroll
    for (int v = 0; v < 8; ++v)
      acc[v] = Cinit[(size_t)(row0 + half * 8 + v) * N + col0 + m];
  }

  const float* arow = A + (size_t)(row0 + m) * K + kk;
  for (int k0 = 0; k0 < K; k0 += 4) {
    v2f a = *(const v2f*)(arow + k0);
    v2f b;
    b.x = panelS[(k0 + kk) * 16 + m];
    b.y = panelS[(k0 + kk + 1) * 16 + m];
    acc = __builtin_amdgcn_wmma_f32_16x16x4_f32(false, a, false, b,
                                                (short)0, acc, false, false);
  }

#pragma unroll
  for (int v = 0; v < 8; ++v)
    C[(size_t)(row0 + half * 8 + v) * N + col0 + m] = acc[v];
}

// ---------------------------------------------------------------------------
// One wave per destination node. Edges (sorted by dst) found by binary search.
// Per chunk of 8 edges:
//   lanes 0..7 compute the 7 geometric features + seq bin,
//   every lane computes 4 of the w[e][k] = leaky(feat @ Ws[idx] + bs[idx], .2),
//   2 groups x 4 V_WMMA_F32_16X16X4_F32: agg[16,64] += w^T(16x4) @ H(4x64).
// Epilogue: agg -> LDS, per-lane GEMV with Wp[1024,O] into cat columns.
// ---------------------------------------------------------------------------
__global__ __launch_bounds__(128) void cdconv_wmma(
    const float* __restrict__ h, const float* __restrict__ pos,
    const float* __restrict__ ori, const int* __restrict__ seq,
    const int* __restrict__ src, const int* __restrict__ dst, int E,
    const float* __restrict__ Ws, const float* __restrict__ bs,
    const float* __restrict__ Wp, float inv_r, int hl,
    float* __restrict__ cat, int colOff, int O, int nNodes) {
  __shared__ float aggS[4][1024];
  __shared__ float featS[4][8][7];
  __shared__ float wS[4][8][16];
  __shared__ int   sS[4][8];
  __shared__ int   idxS[4][8];

  int lane = threadIdx.x & 31;
  int wv   = threadIdx.x >> 5;
  int node = blockIdx.x * 4 + wv;
  if (node >= nNodes) return;            // wave-uniform

  int lo = lower_bound_i32(dst, E, node);
  int hi = lower_bound_i32(dst, E, node + 1);

  float pix = pos[(size_t)node * 3 + 0];
  float piy = pos[(size_t)node * 3 + 1];
  float piz = pos[(size_t)node * 3 + 2];
  float fi[9];
#pragma unroll
  for (int t = 0; t < 9; ++t) fi[t] = ori[(size_t)node * 9 + t];
  int seqi = seq[node];

  v8f acc0 = {}; v8f acc1 = {}; v8f acc2 = {}; v8f acc3 = {};
  int m  = lane & 15;
  int eh = lane >> 4;

  for (int e0 = lo; e0 < hi; e0 += 8) {
    if (lane < 8) {
      int e  = e0 + lane;
      int ec = e < hi ? e : hi - 1;      // clamp: padded edges get w=0 below
      int j  = src[ec];
      sS[wv][lane] = j;
      float dx = pos[(size_t)j * 3 + 0] - pix;
      float dy = pos[(size_t)j * 3 + 1] - piy;
      float dz = pos[(size_t)j * 3 + 2] - piz;
      float dist = sqrtf(dx * dx + dy * dy + dz * dz);
      float inv  = 1.0f / (dist + 1e-9f);
      float ux = dx * inv, uy = dy * inv, uz = dz * inv;
      const float* fj = ori + (size_t)j * 9;
      featS[wv][lane][0] = dist * inv_r;
      featS[wv][lane][1] = fi[0] * ux + fi[1] * uy + fi[2] * uz;
      featS[wv][lane][2] = fi[3] * ux + fi[4] * uy + fi[5] * uz;
      featS[wv][lane][3] = fi[6] * ux + fi[7] * uy + fi[8] * uz;
      featS[wv][lane][4] = fi[0] * fj[0] + fi[1] * fj[1] + fi[2] * fj[2];
      featS[wv][lane][5] = fi[3] * fj[3] + fi[4] * fj[4] + fi[5] * fj[5];
      featS[wv][lane][6] = fi[6] * fj[6] + fi[7] * fj[7] + fi[8] * fj[8];
      int d = seq[j] - seqi;
      d = d < -hl ? -hl : (d > hl ? hl : d);
      idxS[wv][lane] = d + hl;
    }
    wave_lds_sync();

#pragma unroll
    for (int t = 0; t < 4; ++t) {
      int e   = eh + 2 * t;
      int idx = idxS[wv][e];
      const float* f = featS[wv][e];
      float w = bs[idx * 16 + m];
#pragma unroll
      for (int c = 0; c < 7; ++c) w += f[c] * Ws[(idx * 7 + c) * 16 + m];
      w = w > 0.f ? w : 0.2f * w;        // leaky_relu slope 0.2 on edge weights
      if (e0 + e >= hi) w = 0.f;         // zero-pad tail edges
      wS[wv][e][m] = w;
    }
    wave_lds_sync();

#pragma unroll
    for (int g = 0; g < 2; ++g) {
      int kk = g * 4 + (eh << 1);        // this half-wave's K pair in group g
      v2f a;
      a.x = wS[wv][kk][m];
      a.y = wS[wv][kk + 1][m];
      const float* h0 = h + (size_t)sS[wv][kk] * 64;
      const float* h1 = h + (size_t)sS[wv][kk + 1] * 64;
      v2f b;
      b.x = h0[m];      b.y = h1[m];
      acc0 = __builtin_amdgcn_wmma_f32_16x16x4_f32(false, a, false, b, (short)0, acc0, false, false);
      b.x = h0[16 + m]; b.y = h1[16 + m];
      acc1 = __builtin_amdgcn_wmma_f32_16x16x4_f32(false, a, false, b, (short)0, acc1, false, false);
      b.x = h0[32 + m]; b.y = h1[32 + m];
      acc2 = __builtin_amdgcn_wmma_f32_16x16x4_f32(false, a, false, b, (short)0, acc2, false, false);
      b.x = h0[48 + m]; b.y = h1[48 + m];
      acc3 = __builtin_amdgcn_wmma_f32_16x16x4_f32(false, a, false, b, (short)0, acc3, false, false);
    }
    // no tail sync needed: same-wave LDS ops are processed in order, so next
    // iteration's stores cannot pass this iteration's loads.
  }

#pragma unroll
  for (int v = 0; v < 8; ++v) {
    int k = eh * 8 + v;                  // D layout: row = (lane>>4)*8 + v
    aggS[wv][k * 64 +      m] = acc0[v];
    aggS[wv][k * 64 + 16 + m] = acc1[v];
    aggS[wv][k * 64 + 32 + m] = acc2[v];
    aggS[wv][k * 64 + 48 + m] = acc3[v];
  }
  wave_lds_sync();

  if (lane < O) {
    float s = 0.f;
    for (int j2 = 0; j2 < 1024; ++j2) s += aggS[wv][j2] * Wp[(size_t)j2 * O + lane];
    cat[(size_t)node * 64 + colOff + lane] = s;
  }
}

// ---------------------------------------------------------------------------
// Host-side orchestration (graph-capture safe: launches only)
// ---------------------------------------------------------------------------
extern "C" void kernel_launch(void* const* d_in, const int* in_sizes, int n_in,
                              void* d_out, int out_size, void* d_ws, size_t ws_size,
                              hipStream_t stream) {
  (void)n_in; (void)out_size; (void)ws_size;
  constexpr int N = 4096, CIN = 128, COUT = 256, WIDTH = 64;
  constexpr float SLOPE = 0.1f;
  constexpr float R1 = 1.5f * 1.8f, R2 = 1.8f, R3 = 0.75f * 1.8f;

  const float* x      = (const float*)d_in[0];
  const float* pos    = (const float*)d_in[1];
  const float* ori    = (const float*)d_in[2];
  const int*   seq    = (const int*)d_in[3];
  // d_in[4] = batch (unused: edge lists are already per-group)
  const int* src1 = (const int*)d_in[5];  const int* dst1 = (const int*)d_in[6];
  const int* src2 = (const int*)d_in[7];  const int* dst2 = (const int*)d_in[8];
  const int* src3 = (const int*)d_in[9];  const int* dst3 = (const int*)d_in[10];
  int E1 = in_sizes[5], E2 = in_sizes[7], E3 = in_sizes[9];

  const float* id_g = (const float*)d_in[11];
  const float* id_b = (const float*)d_in[12];
  const float* id_W = (const float*)d_in[13];
  const float* in_g = (const float*)d_in[14];
  const float* in_b = (const float*)d_in[15];
  const float* in_W = (const float*)d_in[16];
  const float* mid_g = (const float*)d_in[17];
  const float* mid_b = (const float*)d_in[18];
  const float* Ws1 = (const float*)d_in[19];
  const float* bs1 = (const float*)d_in[20];
  const float* W1  = (const float*)d_in[21];
  const float* Ws2 = (const float*)d_in[22];
  const float* bs2 = (const float*)d_in[23];
  const float* W2  = (const float*)d_in[24];
  const float* Ws3 = (const float*)d_in[25];
  const float* bs3 = (const float*)d_in[26];
  const float* W3  = (const float*)d_in[27];
  const float* out_g = (const float*)d_in[28];
  const float* out_b = (const float*)d_in[29];
  const float* out_W = (const float*)d_in[30];

  // workspace layout (f32)
  float* ws = (float*)d_ws;
  float* meanX = ws;            float* rstdX = ws + 128;
  float* meanH = ws + 256;      float* rstdH = ws + 320;
  float* meanC = ws + 384;      float* rstdC = ws + 448;
  float* a_id     = ws + 512;                         // N*CIN
  float* a_in     = a_id     + (size_t)N * CIN;       // N*CIN
  float* identity = a_in     + (size_t)N * CIN;       // N*COUT
  float* h_pre    = identity + (size_t)N * COUT;      // N*WIDTH
  float* hbuf     = h_pre    + (size_t)N * WIDTH;     // N*WIDTH
  float* catb     = hbuf     + (size_t)N * WIDTH;     // N*WIDTH
  float* a_cat    = catb     + (size_t)N * WIDTH;     // N*WIDTH
  float* out      = (float*)d_out;                    // N*COUT

  // --- input BN stats + activations
  bn_stats_kernel<<<CIN, 256, 0, stream>>>(x, N, CIN, meanX, rstdX);
  bn_act_kernel<<<(N * CIN + 255) / 256, 256, 0, stream>>>(
      x, meanX, rstdX, id_g, id_b, SLOPE, a_id, N * CIN, CIN);
  bn_act_kernel<<<(N * CIN + 255) / 256, 256, 0, stream>>>(
      x, meanX, rstdX, in_g, in_b, SLOPE, a_in, N * CIN, CIN);

  // --- identity and input GEMMs (WMMA f32, async-staged B panels)
  gemm_wmma_f32<<<(N / 64) * (COUT / 16), 128, 0, stream>>>(
      a_id, id_W, nullptr, identity, N, COUT, CIN);
  gemm_wmma_f32<<<(N / 64) * (WIDTH / 16), 128, 0, stream>>>(
      a_in, in_W, nullptr, h_pre, N, WIDTH, CIN);

  // --- mid BN + activation -> h
  bn_stats_kernel<<<WIDTH, 256, 0, stream>>>(h_pre, N, WIDTH, meanH, rstdH);
  bn_act_kernel<<<(N * WIDTH + 255) / 256, 256, 0, stream>>>(
      h_pre, meanH, rstdH, mid_g, mid_b, SLOPE, hbuf, N * WIDTH, WIDTH);

  // --- three CDConv branches write disjoint column blocks of cat [N,64]
  cdconv_wmma<<<N / 4, 128, 0, stream>>>(hbuf, pos, ori, seq, src1, dst1, E1,
      Ws1, bs1, W1, 1.0f / R1, 5 / 2, catb, 0, 32, N);
  cdconv_wmma<<<N / 4, 128, 0, stream>>>(hbuf, pos, ori, seq, src2, dst2, E2,
      Ws2, bs2, W2, 1.0f / R2, 7 / 2, catb, 32, 16, N);
  cdconv_wmma<<<N / 4, 128, 0, stream>>>(hbuf, pos, ori, seq, src3, dst3, E3,
      Ws3, bs3, W3, 1.0f / R3, 11 / 2, catb, 48, 16, N);

  // --- output BN + activation, final GEMM fused with residual (Cinit=identity)
  bn_stats_kernel<<<WIDTH, 256, 0, stream>>>(catb, N, WIDTH, meanC, rstdC);
  bn_act_kernel<<<(N * WIDTH + 255) / 256, 256, 0, stream>>>(
      catb, meanC, rstdC, out_g, out_b, SLOPE, a_cat, N * WIDTH, WIDTH);
  gemm_wmma_f32<<<(N / 64) * (COUT / 16), 128, 0, stream>>>(
      a_cat, out_W, identity, out, N, COUT, WIDTH);
}